// SampleGNN_22548578304211
// MI455X (gfx1250) — compile-verified
//
#include <hip/hip_runtime.h>
#include <hip/hip_bf16.h>
#include <math.h>

// ---------------------------------------------------------------------------
// GCN forward for MI455X (gfx1250, wave32).
//  * Dense GEMMs via V_WMMA_F32_16X16X4_F32 (full f32 precision, RNE FMA).
//    Wave tile 32x64 -> 8 B-loads + 2 ds-loads per 8 WMMAs (balanced issue).
//  * Edge aggregation via dst-CSR gather-reduce (built once, used 4x):
//    no f32 atomics in the layer loop; hw (51MB) is L2-resident (192MB L2),
//    so the random row gather runs at L2 rates, and each agg row is written
//    exactly once instead of 16x atomic-RMW'd.
//  * ReLU fused into the GEMM A-tile staging and the pooling read.
// ---------------------------------------------------------------------------

#define N_NODES 50000
#define N_EDGES 800000
#define N_GRAPHS 512
#define HID 256
#define NODE_FEAT 9
#define LAYERS 4
#define N_PAD 50048            // 64-row padded so every WMMA block is full (EXEC all-1s)
#define ROWS_PER_BLOCK 64
#define SCAN_T 1024

typedef __attribute__((ext_vector_type(2))) float v2f;
typedef __attribute__((ext_vector_type(8))) float v8f;

__device__ inline float4 fma4(float4 a, float s, float4 c) {
    return make_float4(fmaf(a.x, s, c.x), fmaf(a.y, s, c.y),
                       fmaf(a.z, s, c.z), fmaf(a.w, s, c.w));
}

// ---------------- zero fills (deterministic per-call re-init) -------------
__global__ void gcn_zero_f32(float* __restrict__ p, int n) {
    int i = blockIdx.x * 256 + threadIdx.x;
    if (i < n) p[i] = 0.0f;
}
__global__ void gcn_zero_i32(int* __restrict__ p, int n) {
    int i = blockIdx.x * 256 + threadIdx.x;
    if (i < n) p[i] = 0;
}

// ---------------- degree (int, deterministic) / dinv ----------------------
__global__ void gcn_deg_kernel(const int* __restrict__ dst, int* __restrict__ degc, int E) {
    int e = blockIdx.x * 256 + threadIdx.x;
    if (e < E) atomicAdd(&degc[dst[e]], 1);
}
__global__ void gcn_dinv_kernel(const int* __restrict__ degc, float* __restrict__ dinv, int n) {
    int i = blockIdx.x * 256 + threadIdx.x;
    if (i < n) dinv[i] = rsqrtf((float)degc[i] + 1.0f);   // +1 self-loop
}

// ---------------- CSR row offsets: exclusive scan of degc (1 block) -------
__global__ __launch_bounds__(SCAN_T) void gcn_scan_kernel(
    const int* __restrict__ cnt, int* __restrict__ rowoff, int n) {
    __shared__ int sums[SCAN_T];
    const int t = threadIdx.x;
    const int chunk = (n + SCAN_T - 1) / SCAN_T;
    const int begin = t * chunk;
    const int end   = (begin + chunk < n) ? (begin + chunk) : n;
    int s = 0;
    for (int i = begin; i < end; ++i) s += cnt[i];
    sums[t] = s;
    __syncthreads();
    for (int off = 1; off < SCAN_T; off <<= 1) {          // inclusive scan
        int v   = sums[t];
        int add = (t >= off) ? sums[t - off] : 0;
        __syncthreads();
        sums[t] = v + add;
        __syncthreads();
    }
    int prefix = (t == 0) ? 0 : sums[t - 1];              // exclusive chunk prefix
    for (int i = begin; i < end; ++i) {
        rowoff[i] = prefix;
        prefix += cnt[i];
    }
    if (t == SCAN_T - 1) rowoff[n] = prefix;              // == E
}

// ---------------- CSR fill: group edge sources by dst ---------------------
__global__ void gcn_fill_kernel(const int* __restrict__ src, const int* __restrict__ dst,
                                const int* __restrict__ rowoff, int* __restrict__ fillpos,
                                int* __restrict__ csr_src, int E) {
    int e = blockIdx.x * 256 + threadIdx.x;
    if (e < E) {
        int d = dst[e];
        int p = atomicAdd(&fillpos[d], 1);                // int atomics: set is deterministic
        csr_src[rowoff[d] + p] = src[e];
    }
}

// ---------------- embedding: h = relu(x @ embed_W + b), K = 9 -------------
__global__ __launch_bounds__(HID) void gcn_embed_kernel(
    const float* __restrict__ x, const float* __restrict__ W,
    const float* __restrict__ b, float* __restrict__ h) {
    __shared__ float xs[NODE_FEAT];
    const int i = blockIdx.x;          // node
    const int j = threadIdx.x;         // hidden col
    if (j < NODE_FEAT) xs[j] = x[i * NODE_FEAT + j];
    __syncthreads();
    float acc = b[j];
#pragma unroll
    for (int k = 0; k < NODE_FEAT; ++k) acc = fmaf(xs[k], W[k * HID + j], acc);
    h[(size_t)i * HID + j] = fmaxf(acc, 0.0f);
}

// ---------------- dense GEMM: C[N_PAD,256] = relu(A)[N_PAD,256] @ B -------
// Block: 64x256 C tile; relu(A) tile staged in LDS (64KB). 8 waves, each
// wave owns a 32x64 strip: 2 row-tiles x 4 col-tiles, so every B fragment
// feeds two V_WMMA_F32_16X16X4_F32 ops.
__global__ __launch_bounds__(256) void gcn_gemm_wmma(
    const float* __restrict__ A, const float* __restrict__ B, float* __restrict__ C) {
    __shared__ float As[ROWS_PER_BLOCK][HID];          // 64 KB of the 320 KB LDS

    const int tid  = threadIdx.x;
    const int row0 = blockIdx.x * ROWS_PER_BLOCK;

    // Cooperative relu(A)-tile load: 4096 float4, 16/thread, coalesced.
    {
        const float4* Ag  = (const float4*)(A + (size_t)row0 * HID);
        float4*       As4 = (float4*)&As[0][0];
#pragma unroll
        for (int i = 0; i < 16; ++i) {
            float4 v = Ag[tid + i * 256];
            v.x = fmaxf(v.x, 0.0f); v.y = fmaxf(v.y, 0.0f);
            v.z = fmaxf(v.z, 0.0f); v.w = fmaxf(v.w, 0.0f);
            As4[tid + i * 256] = v;
        }
    }
    __syncthreads();

    const int lane = tid & 31;
    const int wave = tid >> 5;         // 0..7
    const int l16  = lane & 15;
    const int lhi  = lane >> 4;        // 0|1 : selects K pair per ISA f32 A/B layout
    const int rgrp = wave >> 2;        // rows rgrp*32 .. +31 (2 row tiles)
    const int cgrp = wave & 3;         // cols cgrp*64 .. +63 (4 col tiles)

    v8f acc[8] = {};                   // [rt*4+t]: 8 C tiles = 64 VGPRs

    for (int k = 0; k < HID; k += 4) {
        const int ka = k + 2 * lhi;
        // A frags (16x4): lane l -> M = l%16, holds K = ka, ka+1
        const v2f a0 = *(const v2f*)&As[rgrp * 32      + l16][ka];
        const v2f a1 = *(const v2f*)&As[rgrp * 32 + 16 + l16][ka];
        const float* Brow0 = B + (size_t)ka * HID;
        const float* Brow1 = Brow0 + HID;
#pragma unroll
        for (int t = 0; t < 4; ++t) {
            const int nn = cgrp * 64 + t * 16 + l16;
            v2f b;
            b.x = Brow0[nn];
            b.y = Brow1[nn];
            acc[t] = __builtin_amdgcn_wmma_f32_16x16x4_f32(
                false, a0, false, b, (short)0, acc[t], false, false);
            acc[4 + t] = __builtin_amdgcn_wmma_f32_16x16x4_f32(
                false, a1, false, b, (short)0, acc[4 + t], false, false);
        }
    }

    // C/D layout: VGPR r, lanes 0-15 -> M=r, lanes 16-31 -> M=r+8; N = lane%16
#pragma unroll
    for (int rt = 0; rt < 2; ++rt) {
#pragma unroll
        for (int t = 0; t < 4; ++t) {
            const int nn = cgrp * 64 + t * 16 + l16;
#pragma unroll
            for (int r = 0; r < 8; ++r) {
                C[(size_t)(row0 + rgrp * 32 + rt * 16 + 8 * lhi + r) * HID + nn] =
                    acc[rt * 4 + t][r];
            }
        }
    }
}

// ---------------- gather-reduce: one wave per dst node --------------------
// agg[d] = hw[d]*dinv[d]^2 + bias + sum_{s in CSR(d)} hw[s] * dinv[s]*dinv[d]
// Lane l owns cols l*8..l*8+7 (two float4) -> 1KB coalesced row reads (L2 hot).
__global__ __launch_bounds__(256) void gcn_gather_kernel(
    const float* __restrict__ hw, const float* __restrict__ dinv,
    const int* __restrict__ rowoff, const int* __restrict__ csr_src,
    const float* __restrict__ bias, float* __restrict__ agg, int n) {
    const int d = blockIdx.x * 8 + (threadIdx.x >> 5);
    if (d >= n) return;
    const int lane = threadIdx.x & 31;
    const int c0 = lane * 8;
    const float dd = dinv[d];

    float4 a0, a1;
    {   // self-loop + bias base (deterministic)
        const float  sc = dd * dd;
        const float4 h0 = *(const float4*)(hw + (size_t)d * HID + c0);
        const float4 h1 = *(const float4*)(hw + (size_t)d * HID + c0 + 4);
        const float4 b0 = *(const float4*)(bias + c0);
        const float4 b1 = *(const float4*)(bias + c0 + 4);
        a0 = fma4(h0, sc, b0);
        a1 = fma4(h1, sc, b1);
    }
    const int e0 = rowoff[d], e1 = rowoff[d + 1];
    for (int e = e0; e < e1; ++e) {
        const int   s = csr_src[e];
        const float w = dinv[s] * dd;
        const float4 m0 = *(const float4*)(hw + (size_t)s * HID + c0);
        const float4 m1 = *(const float4*)(hw + (size_t)s * HID + c0 + 4);
        a0 = fma4(m0, w, a0);
        a1 = fma4(m1, w, a1);
    }
    *(float4*)(agg + (size_t)d * HID + c0)     = a0;
    *(float4*)(agg + (size_t)d * HID + c0 + 4) = a1;
}

// ---------------- pooling: per-graph segment sums of relu(agg) ------------
__global__ __launch_bounds__(HID) void gcn_pool_kernel(
    const float* __restrict__ agg, const int* __restrict__ batch,
    float* __restrict__ gsum, float* __restrict__ gcnt) {
    const int i = blockIdx.x, j = threadIdx.x;
    const int g = batch[i];
    atomicAdd(&gsum[(size_t)g * HID + j], fmaxf(agg[(size_t)i * HID + j], 0.0f));
    if (j == 0) atomicAdd(&gcnt[g], 1.0f);   // +1.0 adds exact -> order independent
}

// ---------------- head: out[g] = (gsum[g]/cnt) . out_W + out_b ------------
__global__ __launch_bounds__(HID) void gcn_out_kernel(
    const float* __restrict__ gsum, const float* __restrict__ gcnt,
    const float* __restrict__ W, const float* __restrict__ b,
    float* __restrict__ out) {
    __shared__ float red[8];
    const int g = blockIdx.x, j = threadIdx.x;
    const float cnt = fmaxf(gcnt[g], 1.0f);
    float v = (gsum[(size_t)g * HID + j] / cnt) * W[j];
#pragma unroll
    for (int off = 16; off > 0; off >>= 1) v += __shfl_down(v, off, 32);
    if ((j & 31) == 0) red[j >> 5] = v;
    __syncthreads();
    if (j == 0) {
        float s = 0.0f;
#pragma unroll
        for (int w = 0; w < 8; ++w) s += red[w];
        out[g] = s + b[0];
    }
}

// ---------------------------------------------------------------------------
extern "C" void kernel_launch(void* const* d_in, const int* in_sizes, int n_in,
                              void* d_out, int out_size, void* d_ws, size_t ws_size,
                              hipStream_t stream) {
    const float* x     = (const float*)d_in[0];   // [N,9]
    const int*   ei    = (const int*)  d_in[1];   // [2,E]: row0=src, row1=dst
    const int*   batch = (const int*)  d_in[2];   // [N]
    const float* embW  = (const float*)d_in[3];   // [9,256]
    const float* embB  = (const float*)d_in[4];   // [256]
    const float* convW = (const float*)d_in[5];   // [4,256,256]
    const float* convB = (const float*)d_in[6];   // [4,256]
    const float* outW  = (const float*)d_in[7];   // [256,1]
    const float* outB  = (const float*)d_in[8];   // [1]
    float* out = (float*)d_out;                   // [512]

    const int N = N_NODES, E = N_EDGES, G = N_GRAPHS;
    const int* src = ei;
    const int* dst = ei + E;

    // Workspace layout (~158 MB):
    float* h       = (float*)d_ws;                    // [N_PAD,256]
    float* hw      = h    + (size_t)N_PAD * HID;      // [N_PAD,256]
    float* agg     = hw   + (size_t)N_PAD * HID;      // [N_PAD,256]
    float* dinv    = agg  + (size_t)N_PAD * HID;      // [N]
    float* gsum    = dinv + N;                        // [G,256]
    float* gcnt    = gsum + (size_t)G * HID;          // [G]
    int*   degc    = (int*)(gcnt + G);                // [N]
    int*   rowoff  = degc + N;                        // [N+1]
    int*   fillpos = rowoff + (N + 1);                // [N]
    int*   csr_src = fillpos + N;                     // [E]

    // Deterministic per-call re-init.
    gcn_zero_i32<<<(N + 255) / 256, 256, 0, stream>>>(degc, N);
    gcn_zero_i32<<<(N + 255) / 256, 256, 0, stream>>>(fillpos, N);
    gcn_zero_f32<<<(G * HID + 255) / 256, 256, 0, stream>>>(gsum, G * HID);
    gcn_zero_f32<<<(G + 255) / 256, 256, 0, stream>>>(gcnt, G);
    gcn_zero_f32<<<((N_PAD - N) * HID + 255) / 256, 256, 0, stream>>>(
        h + (size_t)N * HID, (N_PAD - N) * HID);
    gcn_zero_f32<<<((N_PAD - N) * HID + 255) / 256, 256, 0, stream>>>(
        agg + (size_t)N * HID, (N_PAD - N) * HID);

    // Degree / normalization / CSR (edge structure constant across layers).
    gcn_deg_kernel<<<(E + 255) / 256, 256, 0, stream>>>(dst, degc, E);
    gcn_dinv_kernel<<<(N + 255) / 256, 256, 0, stream>>>(degc, dinv, N);
    gcn_scan_kernel<<<1, SCAN_T, 0, stream>>>(degc, rowoff, N);
    gcn_fill_kernel<<<(E + 255) / 256, 256, 0, stream>>>(src, dst, rowoff, fillpos,
                                                         csr_src, E);

    gcn_embed_kernel<<<N, HID, 0, stream>>>(x, embW, embB, h);

    for (int l = 0; l < LAYERS; ++l) {
        const float* Ain = (l == 0) ? h : agg;        // relu fused into A-tile staging
        gcn_gemm_wmma<<<N_PAD / ROWS_PER_BLOCK, 256, 0, stream>>>(
            Ain, convW + (size_t)l * HID * HID, hw);
        gcn_gather_kernel<<<(N + 7) / 8, 256, 0, stream>>>(
            hw, dinv, rowoff, csr_src, convB + (size_t)l * HID, agg, N);
    }

    gcn_pool_kernel<<<N, HID, 0, stream>>>(agg, batch, gsum, gcnt);
    gcn_out_kernel<<<G, HID, 0, stream>>>(gsum, gcnt, outW, outB, out);
}